// DiscriminativeLoss_59828894433896
// MI455X (gfx1250) — compile-verified
//
#include <hip/hip_runtime.h>
#include <hip/hip_bf16.h>

typedef __attribute__((ext_vector_type(2))) float v2f;
typedef __attribute__((ext_vector_type(8))) float v8f;

#define KMAX 128
#define DDIM 16

// ws float layout:
//  [0,            KMAX*DDIM)           sums        (2048)
//  [KMAX*DDIM,    KMAX*DDIM+KMAX)      counts      (128)
//  [+KMAX,        +2*KMAX)             var_sum     (128)
//  [+2*KMAX,      +2*KMAX+KMAX*DDIM)   means       (2048)
#define WS_SUMS   0
#define WS_COUNTS (KMAX*DDIM)
#define WS_VAR    (KMAX*DDIM + KMAX)
#define WS_MEANS  (KMAX*DDIM + 2*KMAX)
#define WS_TOTAL  (2*KMAX*DDIM + 2*KMAX)

__global__ void dl_init(float* __restrict__ ws) {
  int i = blockIdx.x * blockDim.x + threadIdx.x;
  if (i < WS_MEANS) ws[i] = 0.0f;   // zero sums, counts, var_sum
}

// Pass 1: segment sums + counts, privatized in LDS, one atomic flush per block.
__global__ void dl_segsum(const float4* __restrict__ emb4,
                          const int* __restrict__ labels,
                          float* __restrict__ ws, int N) {
  __shared__ float ls[KMAX * DDIM];
  __shared__ float lc[KMAX];
  int t = threadIdx.x;
  for (int i = t; i < KMAX * DDIM; i += blockDim.x) ls[i] = 0.0f;
  if (t < KMAX) lc[t] = 0.0f;
  __syncthreads();

  int stride = gridDim.x * blockDim.x;
  for (int i = blockIdx.x * blockDim.x + t; i < N; i += stride) {
    int lbl = labels[i] & (KMAX - 1);
    float4 e0 = emb4[i * 4 + 0];
    float4 e1 = emb4[i * 4 + 1];
    float4 e2 = emb4[i * 4 + 2];
    float4 e3 = emb4[i * 4 + 3];
    float* b = &ls[lbl * DDIM];
    atomicAdd(&b[0],  e0.x); atomicAdd(&b[1],  e0.y);
    atomicAdd(&b[2],  e0.z); atomicAdd(&b[3],  e0.w);
    atomicAdd(&b[4],  e1.x); atomicAdd(&b[5],  e1.y);
    atomicAdd(&b[6],  e1.z); atomicAdd(&b[7],  e1.w);
    atomicAdd(&b[8],  e2.x); atomicAdd(&b[9],  e2.y);
    atomicAdd(&b[10], e2.z); atomicAdd(&b[11], e2.w);
    atomicAdd(&b[12], e3.x); atomicAdd(&b[13], e3.y);
    atomicAdd(&b[14], e3.z); atomicAdd(&b[15], e3.w);
    atomicAdd(&lc[lbl], 1.0f);
  }
  __syncthreads();
  for (int i = t; i < KMAX * DDIM; i += blockDim.x)
    if (ls[i] != 0.0f) atomicAdd(&ws[WS_SUMS + i], ls[i]);
  if (t < KMAX && lc[t] != 0.0f) atomicAdd(&ws[WS_COUNTS + t], lc[t]);
}

__global__ void dl_means(float* __restrict__ ws, const int* __restrict__ kptr) {
  int K = *kptr; if (K > KMAX) K = KMAX;
  for (int i = threadIdx.x; i < K * DDIM; i += blockDim.x) {
    int k = i / DDIM;
    ws[WS_MEANS + i] = ws[WS_SUMS + i] / fmaxf(ws[WS_COUNTS + k], 1.0f);
  }
}

// Pass 2: pull (variance) term. Means cached in LDS transposed (bank-conflict free gather).
__global__ void dl_var(const float4* __restrict__ emb4,
                       const int* __restrict__ labels,
                       float* __restrict__ ws, int N) {
  __shared__ float mT[DDIM * KMAX];   // mT[d*KMAX + k]
  __shared__ float vs[KMAX];
  int t = threadIdx.x;
  for (int i = t; i < KMAX * DDIM; i += blockDim.x) {
    int k = i / DDIM, d = i % DDIM;
    mT[d * KMAX + k] = ws[WS_MEANS + i];
  }
  if (t < KMAX) vs[t] = 0.0f;
  __syncthreads();

  int stride = gridDim.x * blockDim.x;
  for (int i = blockIdx.x * blockDim.x + t; i < N; i += stride) {
    int lbl = labels[i] & (KMAX - 1);
    float4 e0 = emb4[i * 4 + 0];
    float4 e1 = emb4[i * 4 + 1];
    float4 e2 = emb4[i * 4 + 2];
    float4 e3 = emb4[i * 4 + 3];
    float e[DDIM] = {e0.x,e0.y,e0.z,e0.w, e1.x,e1.y,e1.z,e1.w,
                     e2.x,e2.y,e2.z,e2.w, e3.x,e3.y,e3.z,e3.w};
    float sq = 0.0f;
    #pragma unroll
    for (int d = 0; d < DDIM; ++d) {
      float df = e[d] - mT[d * KMAX + lbl];
      sq += df * df;
    }
    float dist = sqrtf(sq);                  // safe_sqrt: sqrtf(0)==0
    float h = fmaxf(dist - 0.5f, 0.0f);      // DELTA_V
    atomicAdd(&vs[lbl], h * h);
  }
  __syncthreads();
  if (t < KMAX && vs[t] != 0.0f) atomicAdd(&ws[WS_VAR + t], vs[t]);
}

// Final: WMMA Gram matrix of means -> push term; + var/reg terms; block reduce -> out.
__global__ void dl_final(const float* __restrict__ ws, float* __restrict__ out,
                         const int* __restrict__ kptr) {
  __shared__ float mL[KMAX * DDIM];   // row-major (k, d)
  __shared__ float norm2[KMAX];
  __shared__ float red[256];
  int K = *kptr; if (K > KMAX) K = KMAX;
  int t = threadIdx.x;

  for (int i = t; i < KMAX * DDIM; i += blockDim.x)
    mL[i] = (i < K * DDIM) ? ws[WS_MEANS + i] : 0.0f;
  __syncthreads();
  if (t < KMAX) {
    float s = 0.0f;
    #pragma unroll
    for (int d = 0; d < DDIM; ++d) { float m = mL[t * DDIM + d]; s += m * m; }
    norm2[t] = s;
  }
  __syncthreads();

  // --- push term via V_WMMA_F32_16X16X4_F32: G = M * M^T, upper-triangle tiles ---
  int wave = t >> 5;
  int lane = t & 31;
  int half = lane >> 4;      // 0: lanes 0-15, 1: lanes 16-31
  int m16  = lane & 15;
  int KT = K >> 4;           // 16-wide tiles per dim (8 for K=128)
  float dpart = 0.0f;

  for (int tr = wave; tr < KT; tr += 8) {          // wave-uniform: EXEC all-1s
    int I0 = tr * 16;
    for (int tc = tr; tc < KT; ++tc) {             // upper-triangle tiles only
      int J0 = tc * 16;
      v8f acc = {};
      #pragma unroll
      for (int k0 = 0; k0 < DDIM; k0 += 4) {
        // A: 16x4 (rows I0..I0+15). lanes0-15 -> K=k0+{0,1}; lanes16-31 -> K=k0+{2,3}
        v2f a, b;
        a.x = mL[(I0 + m16) * DDIM + k0 + 2 * half + 0];
        a.y = mL[(I0 + m16) * DDIM + k0 + 2 * half + 1];
        // B: 4x16 = (M rows J0..J0+15)^T, mirrored layout
        b.x = mL[(J0 + m16) * DDIM + k0 + 2 * half + 0];
        b.y = mL[(J0 + m16) * DDIM + k0 + 2 * half + 1];
        acc = __builtin_amdgcn_wmma_f32_16x16x4_f32(
            /*neg_a=*/false, a, /*neg_b=*/false, b,
            /*c_mod=*/(short)0, acc, /*reuse_a=*/false, /*reuse_b=*/false);
      }
      // C/D layout: VGPR v, lane -> row = v + 8*half, col = m16
      #pragma unroll
      for (int v = 0; v < 8; ++v) {
        int i = I0 + v + 8 * half;
        int j = J0 + m16;
        if (j > i && i < K && j < K) {
          float d2 = norm2[i] + norm2[j] - 2.0f * acc[v];
          float cd = sqrtf(fmaxf(d2, 0.0f));
          float h  = fmaxf(3.0f - cd, 0.0f);       // 2*DELTA_D
          dpart += h * h;
        }
      }
    }
  }

  // --- combine: var_loss + dist_loss + GAMMA*reg_loss ---
  float invK = 1.0f / (float)K;
  int num_pairs = (K * (K - 1)) / 2; if (num_pairs < 1) num_pairs = 1;
  float contrib = dpart / (float)num_pairs;
  if (t < K) {
    float c = fmaxf(ws[WS_COUNTS + t], 1.0f);
    contrib += (ws[WS_VAR + t] / c) * invK;                 // ALPHA=1
    contrib += 0.001f * sqrtf(norm2[t]) * invK;             // GAMMA
  }
  red[t] = contrib;
  __syncthreads();
  for (int s = 128; s > 0; s >>= 1) {
    if (t < s) red[t] += red[t + s];
    __syncthreads();
  }
  if (t == 0) out[0] = red[0];
}

extern "C" void kernel_launch(void* const* d_in, const int* in_sizes, int n_in,
                              void* d_out, int out_size, void* d_ws, size_t ws_size,
                              hipStream_t stream) {
  const float* emb   = (const float*)d_in[0];
  const int* labels  = (const int*)d_in[1];
  const int* kptr    = (const int*)d_in[2];
  float* out = (float*)d_out;
  float* ws  = (float*)d_ws;
  int N = in_sizes[0] / DDIM;

  int blocks = (N + 255) / 256;
  if (blocks > 1024) blocks = 1024;
  if (blocks < 1) blocks = 1;

  dl_init  <<<(WS_MEANS + 255) / 256, 256, 0, stream>>>(ws);
  dl_segsum<<<blocks, 256, 0, stream>>>((const float4*)emb, labels, ws, N);
  dl_means <<<1, 256, 0, stream>>>(ws, kptr);
  dl_var   <<<blocks, 256, 0, stream>>>((const float4*)emb, labels, ws, N);
  dl_final <<<1, 256, 0, stream>>>(ws, out, kptr);
}